// Conv1D_LowRankLSTM_Attention_89876485636761
// MI455X (gfx1250) — compile-verified
//
#include <hip/hip_runtime.h>
#include <math.h>

// ---------------------------------------------------------------------------
// Conv1D (4 depthwise+pointwise+BN+ReLU blocks) -> 2-layer low-rank LSTM
// -> FC.  fp32 throughout; matrix work on V_WMMA_F32_16X16X4_F32 (CDNA5).
// ---------------------------------------------------------------------------

typedef float v2f __attribute__((ext_vector_type(2)));
typedef float v8f __attribute__((ext_vector_type(8)));

namespace {
constexpr int kB    = 256;   // batch
constexpr int kL0   = 1024;  // input length
constexpr int kCIN  = 16;
constexpr int kHID  = 512;
constexpr int kNC   = 10;
constexpr int kT    = 1032;  // L0 + 8 after four conv blocks
constexpr int kG    = 4 * kHID;  // 2048 gate width
}

// ---------------------------------------------------------------------------
// Generic fp32 WMMA GEMM:  C[M,N] = A[M,K] @ B[K,N]  (+ bias[N] if non-null)
// Row-major, M%128==0, N%64==0, K%32==0.  grid = (N/64, M/128), block = 256.
// Block tile 128x64; each wave32 owns a 16x64 strip (4 x v8f accumulators).
// ---------------------------------------------------------------------------
#define LDS_AS 36   // padded row stride (floats) of A tile (128 x 32); %4==0
#define LDS_BP 96   // stride (floats) per K-pair of B tile (16 pairs x 64x2)

__global__ __launch_bounds__(256) void gemm_wmma_f32(
    const float* __restrict__ A, int lda,
    const float* __restrict__ Bm, int ldb,
    float* __restrict__ C, int ldc,
    int M, int N, int K,
    const float* __restrict__ bias)
{
  __shared__ __align__(16) float As[128 * LDS_AS];        // 18432 B
  __shared__ __align__(16) float Bs[16 * LDS_BP];         //  6144 B

  const int tid  = threadIdx.x;
  const int lane = tid & 31;
  const int wave = tid >> 5;       // 0..7  -> rows [wave*16, wave*16+16)
  const int half = lane >> 4;      // 0 / 1  (K-half of fragment)
  const int lrow = lane & 15;

  const int mBase = blockIdx.y * 128;
  const int nBase = blockIdx.x * 64;

  // A staging: 128x32 tile, 16 floats per thread (one row half per thread)
  const int arow = tid >> 1;               // 0..127
  const int acol = (tid & 1) * 16;         // 0 or 16
  // B staging: rows (2r, 2r+1), 4 cols per thread, interleaved K-pairs
  const int brp  = tid >> 4;               // K-pair index 0..15
  const int bcol = (tid & 15) * 4;         // 0..60

  v8f acc0 = {};
  v8f acc1 = {};
  v8f acc2 = {};
  v8f acc3 = {};

  for (int kb = 0; kb < K; kb += 32) {
    // ---- stage A (128x32) -------------------------------------------------
    const float* ga = A + (size_t)(mBase + arow) * lda + kb + acol;
    float4 av0 = *(const float4*)(ga);
    float4 av1 = *(const float4*)(ga + 4);
    float4 av2 = *(const float4*)(ga + 8);
    float4 av3 = *(const float4*)(ga + 12);
    float* asr = &As[arow * LDS_AS + acol];
    *(float4*)(asr)      = av0;
    *(float4*)(asr + 4)  = av1;
    *(float4*)(asr + 8)  = av2;
    *(float4*)(asr + 12) = av3;

    // ---- stage B (32x64) as interleaved K-pairs ---------------------------
    const float* gb = Bm + (size_t)(kb + 2 * brp) * ldb + nBase + bcol;
    float4 blo = *(const float4*)(gb);
    float4 bhi = *(const float4*)(gb + ldb);
    float* bsr = &Bs[brp * LDS_BP + bcol * 2];
    ((v2f*)bsr)[0] = (v2f){blo.x, bhi.x};
    ((v2f*)bsr)[1] = (v2f){blo.y, bhi.y};
    ((v2f*)bsr)[2] = (v2f){blo.z, bhi.z};
    ((v2f*)bsr)[3] = (v2f){blo.w, bhi.w};

    if (kb + 32 < K) {                     // hint next K chunk into cache
      __builtin_prefetch(ga + 32, 0, 1);
      __builtin_prefetch(gb + (size_t)32 * ldb, 0, 1);
    }
    __syncthreads();

    // ---- 32 WMMAs per wave per K chunk (1 A frag feeds 4 WMMAs) -----------
    #pragma unroll
    for (int k4 = 0; k4 < 32; k4 += 4) {
      const int kk = k4 + 2 * half;        // even
      // A fragment: lane (half,lrow) holds A[M=lrow][K=kk], A[M=lrow][K=kk+1]
      v2f afr = *(const v2f*)(&As[(wave * 16 + lrow) * LDS_AS + kk]);
      // B fragments: single b64 each -> {B[kk][n], B[kk+1][n]}
      const float* bp = &Bs[(kk >> 1) * LDS_BP + lrow * 2];
      v2f b0 = *(const v2f*)(bp);
      v2f b1 = *(const v2f*)(bp + 32);
      v2f b2 = *(const v2f*)(bp + 64);
      v2f b3 = *(const v2f*)(bp + 96 - LDS_BP + LDS_BP); // = bp + 96
      acc0 = __builtin_amdgcn_wmma_f32_16x16x4_f32(false, afr, false, b0,
                                                   (short)0, acc0, false, false);
      acc1 = __builtin_amdgcn_wmma_f32_16x16x4_f32(false, afr, false, b1,
                                                   (short)0, acc1, false, false);
      acc2 = __builtin_amdgcn_wmma_f32_16x16x4_f32(false, afr, false, b2,
                                                   (short)0, acc2, false, false);
      acc3 = __builtin_amdgcn_wmma_f32_16x16x4_f32(false, afr, false, b3,
                                                   (short)0, acc3, false, false);
    }
    __syncthreads();
  }

  // ---- epilogue: C/D layout lane(half,lrow) vgpr v -> row v+8*half, col lrow
  const int col0 = nBase + lrow;
  float bias0 = 0.f, bias1 = 0.f, bias2 = 0.f, bias3 = 0.f;
  if (bias) {
    bias0 = bias[col0];
    bias1 = bias[col0 + 16];
    bias2 = bias[col0 + 32];
    bias3 = bias[col0 + 48];
  }
  #pragma unroll
  for (int v = 0; v < 8; ++v) {
    const int row = mBase + wave * 16 + v + 8 * half;
    float* cr = C + (size_t)row * ldc + col0;
    cr[0]  = acc0[v] + bias0;
    cr[16] = acc1[v] + bias1;
    cr[32] = acc2[v] + bias2;
    cr[48] = acc3[v] + bias3;
  }
}

// ---------------------------------------------------------------------------
// Conv-block kernels (tiny fraction of total FLOPs -> plain VALU)
// ---------------------------------------------------------------------------
__global__ void dwconv_k(const float* __restrict__ in,
                         int sB, int sC, int sL, int Lin,
                         const float* __restrict__ wk,
                         const float* __restrict__ wb,
                         float* __restrict__ out, int Lout, int C)
{
  const int idx = blockIdx.x * blockDim.x + threadIdx.x;
  const int total = kB * C * Lout;
  if (idx >= total) return;
  const int l = idx % Lout;
  const int c = (idx / Lout) % C;
  const int b = idx / (Lout * C);
  float s = wb[c];
  const float* ip = in + (size_t)b * sB + (size_t)c * sC;
  #pragma unroll
  for (int k = 0; k < 7; ++k) {
    const int li = l - 4 + k;
    if (li >= 0 && li < Lin) s += ip[(size_t)li * sL] * wk[c * 7 + k];
  }
  out[idx] = s;   // (B, C, Lout) contiguous
}

__global__ __launch_bounds__(256) void pwconv_stats_k(
    const float* __restrict__ in,   // (B, Cin, L)
    const float* __restrict__ pwk,  // (32, Cin)
    const float* __restrict__ pwb,  // (32)
    float* __restrict__ out,        // (B, 32, L)
    float* __restrict__ sums,       // [0..31]=sum, [32..63]=sumsq
    int Cin, int L)
{
  const int l  = blockIdx.x * 256 + threadIdx.x;
  const int co = blockIdx.y;
  const int b  = blockIdx.z;
  float v = 0.f;
  if (l < L) {
    const float* ip = in + (size_t)b * Cin * L + l;
    float acc = pwb[co];
    for (int ci = 0; ci < Cin; ++ci) acc += ip[(size_t)ci * L] * pwk[co * Cin + ci];
    out[((size_t)b * 32 + co) * L + l] = acc;
    v = acc;
  }
  __shared__ float r1[256];
  __shared__ float r2[256];
  r1[threadIdx.x] = v;
  r2[threadIdx.x] = v * v;
  __syncthreads();
  for (int s = 128; s > 0; s >>= 1) {
    if (threadIdx.x < s) {
      r1[threadIdx.x] += r1[threadIdx.x + s];
      r2[threadIdx.x] += r2[threadIdx.x + s];
    }
    __syncthreads();
  }
  if (threadIdx.x == 0) {
    atomicAdd(&sums[co], r1[0]);
    atomicAdd(&sums[32 + co], r2[0]);
  }
}

__global__ void bn_finalize_k(const float* __restrict__ sums,
                              const float* __restrict__ g,
                              const float* __restrict__ bb,
                              float* __restrict__ scale,
                              float* __restrict__ shift, float invN)
{
  const int c = threadIdx.x;
  if (c < 32) {
    const float m   = sums[c] * invN;
    const float var = sums[32 + c] * invN - m * m;
    const float sc  = g[c] * rsqrtf(var + 1e-5f);
    scale[c] = sc;
    shift[c] = bb[c] - m * sc;
  }
}

__global__ void bnrelu_k(const float* __restrict__ in,
                         const float* __restrict__ scale,
                         const float* __restrict__ shift,
                         float* __restrict__ out, int L, int toTBC)
{
  const int idx = blockIdx.x * blockDim.x + threadIdx.x;
  const int total = kB * 32 * L;
  if (idx >= total) return;
  const int l = idx % L;
  const int c = (idx / L) % 32;
  const int b = idx / (L * 32);
  float v = in[idx] * scale[c] + shift[c];
  v = fmaxf(v, 0.f);
  if (toTBC) out[((size_t)l * kB + b) * 32 + c] = v;   // (T, B, 32)
  else       out[idx] = v;                              // (B, 32, L) in-place
}

// ---------------------------------------------------------------------------
// LSTM cell elementwise update (gates laid out [i | f | g | o], width 2048)
// ---------------------------------------------------------------------------
__global__ void lstm_update_k(const float* __restrict__ gates,
                              float* __restrict__ h, float* __restrict__ c)
{
  const int idx = blockIdx.x * blockDim.x + threadIdx.x;
  if (idx >= kB * kHID) return;
  const int b = idx >> 9;      // /512
  const int j = idx & 511;
  const float* g = gates + (size_t)b * kG;
  const float gi = g[j];
  const float gf = g[kHID + j];
  const float gg = g[2 * kHID + j];
  const float go = g[3 * kHID + j];
  const float si = 1.f / (1.f + expf(-gi));
  const float sf = 1.f / (1.f + expf(-gf));
  const float so = 1.f / (1.f + expf(-go));
  const float cn = sf * c[idx] + si * tanhf(gg);
  c[idx] = cn;
  h[idx] = so * tanhf(cn);
}

// ---------------------------------------------------------------------------
// Prepack / utility kernels
// ---------------------------------------------------------------------------
__global__ void catrows_k(const float* __restrict__ top,
                          const float* __restrict__ bot,
                          float* __restrict__ out, int rows, int cols)
{
  const int idx = blockIdx.x * blockDim.x + threadIdx.x;
  if (idx >= 2 * rows * cols) return;
  const int r  = idx / cols;
  const int cc = idx % cols;
  out[idx] = (r < rows) ? top[(size_t)r * cols + cc]
                        : bot[(size_t)(r - rows) * cols + cc];
}

__global__ void addvec_k(const float* __restrict__ a,
                         const float* __restrict__ b,
                         float* __restrict__ out, int n)
{
  const int idx = blockIdx.x * blockDim.x + threadIdx.x;
  if (idx < n) out[idx] = a[idx] + b[idx];
}

__global__ void zero_k(float* __restrict__ p, int n)
{
  const int idx = blockIdx.x * blockDim.x + threadIdx.x;
  if (idx < n) p[idx] = 0.f;
}

__global__ void fc_k(const float* __restrict__ h,
                     const float* __restrict__ w,   // (NC, HID)
                     const float* __restrict__ bb,
                     float* __restrict__ out)
{
  const int idx = blockIdx.x * blockDim.x + threadIdx.x;
  if (idx >= kB * kNC) return;
  const int b = idx / kNC;
  const int n = idx % kNC;
  float s = bb[n];
  const float* hp = h + (size_t)b * kHID;
  const float* wp = w + (size_t)n * kHID;
  for (int j = 0; j < kHID; ++j) s += hp[j] * wp[j];
  out[idx] = s;
}

// ---------------------------------------------------------------------------
static inline int cdiv(int a, int b) { return (a + b - 1) / b; }

extern "C" void kernel_launch(void* const* d_in, const int* in_sizes, int n_in,
                              void* d_out, int out_size, void* d_ws, size_t ws_size,
                              hipStream_t stream)
{
  (void)in_sizes; (void)n_in; (void)out_size; (void)ws_size;

  const float* x = (const float*)d_in[0];
  const float *dwk[4], *dwb[4], *pwk[4], *pwb[4], *bng[4], *bnb[4];
  for (int i = 0; i < 4; ++i) {
    dwk[i] = (const float*)d_in[1 + 6 * i];
    dwb[i] = (const float*)d_in[2 + 6 * i];
    pwk[i] = (const float*)d_in[3 + 6 * i];
    pwb[i] = (const float*)d_in[4 + 6 * i];
    bng[i] = (const float*)d_in[5 + 6 * i];
    bnb[i] = (const float*)d_in[6 + 6 * i];
  }
  const float* Ui1  = (const float*)d_in[25];
  const float* Vi1  = (const float*)d_in[26];
  const float* Uh1  = (const float*)d_in[27];
  const float* Vh1  = (const float*)d_in[28];
  const float* bi1  = (const float*)d_in[29];
  const float* bh1  = (const float*)d_in[30];
  const float* Ui2  = (const float*)d_in[31];
  const float* Vi2  = (const float*)d_in[32];
  const float* Uh2  = (const float*)d_in[33];
  const float* Vh2  = (const float*)d_in[34];
  const float* bi2  = (const float*)d_in[35];
  const float* bh2  = (const float*)d_in[36];
  const float* fcw  = (const float*)d_in[37];
  const float* fcb  = (const float*)d_in[38];

  // ---- workspace layout (floats); total ~110 MB ---------------------------
  float* ws = (float*)d_ws;
  size_t o = 0;
  const size_t CONVSZ = (size_t)kB * 32 * kT;   // 8,454,144 floats
  float* convA = ws + o; o += CONVSZ;           // also aliased as Xall (T,B,32)
  float* convB = ws + o; o += CONVSZ;
  float* dwT   = ws + o; o += CONVSZ;
  float* V1cat = ws + o; o += (size_t)256 * kG;
  float* V2cat = ws + o; o += (size_t)256 * kG;
  float* b1cat = ws + o; o += kG;
  float* b2cat = ws + o; o += kG;
  float* h1 = ws + o; o += (size_t)kB * kHID;
  float* c1 = ws + o; o += (size_t)kB * kHID;
  float* h2 = ws + o; o += (size_t)kB * kHID;
  float* c2 = ws + o; o += (size_t)kB * kHID;
  float* P1 = ws + o; o += (size_t)kB * 256;
  float* P2 = ws + o; o += (size_t)kB * 256;
  float* gates = ws + o; o += (size_t)kB * kG;
  float* stats = ws + o; o += 64;
  float* bnsc  = ws + o; o += 32;
  float* bnsh  = ws + o; o += 32;
  float* Xall  = convA;

  // ---- init h/c (contiguous block) and prepack [Vi;Vh], bi+bh -------------
  zero_k<<<cdiv(4 * kB * kHID, 256), 256, 0, stream>>>(h1, 4 * kB * kHID);
  catrows_k<<<cdiv(256 * kG, 256), 256, 0, stream>>>(Vi1, Vh1, V1cat, 128, kG);
  catrows_k<<<cdiv(256 * kG, 256), 256, 0, stream>>>(Vi2, Vh2, V2cat, 128, kG);
  addvec_k<<<cdiv(kG, 256), 256, 0, stream>>>(bi1, bh1, b1cat, kG);
  addvec_k<<<cdiv(kG, 256), 256, 0, stream>>>(bi2, bh2, b2cat, kG);

  // ---- conv blocks --------------------------------------------------------
  const int Lin[4]  = {1024, 1026, 1028, 1030};
  const int Lout[4] = {1026, 1028, 1030, 1032};
  const int Cin[4]  = {16, 32, 32, 32};
  const float* bin[4] = {x, convA, convB, convA};
  float* pout[4] = {convA, convB, convA, convB};

  for (int i = 0; i < 4; ++i) {
    zero_k<<<1, 64, 0, stream>>>(stats, 64);
    int sB, sC, sL;
    if (i == 0) { sB = kL0 * kCIN; sC = 1; sL = kCIN; }       // x: (B,L,CIN)
    else        { sB = 32 * Lin[i]; sC = Lin[i]; sL = 1; }    // (B,32,L)
    const int totDW = kB * Cin[i] * Lout[i];
    dwconv_k<<<cdiv(totDW, 256), 256, 0, stream>>>(
        bin[i], sB, sC, sL, Lin[i], dwk[i], dwb[i], dwT, Lout[i], Cin[i]);
    dim3 pg(cdiv(Lout[i], 256), 32, kB);
    pwconv_stats_k<<<pg, 256, 0, stream>>>(
        dwT, pwk[i], pwb[i], pout[i], stats, Cin[i], Lout[i]);
    bn_finalize_k<<<1, 32, 0, stream>>>(
        stats, bng[i], bnb[i], bnsc, bnsh, 1.f / (float)(kB * Lout[i]));
    const int totBN = kB * 32 * Lout[i];
    bnrelu_k<<<cdiv(totBN, 256), 256, 0, stream>>>(
        pout[i], bnsc, bnsh, (i == 3) ? Xall : pout[i], Lout[i], (i == 3) ? 1 : 0);
  }

  // ---- LSTM scan: 1032 steps, all matrix work on fp32 WMMA ----------------
  const dim3 blk(256);
  const dim3 g_s1(128 / 64, kB / 128);   // stage-1: M=256, N=128
  const dim3 g_s2(kG / 64, kB / 128);    // gates:   M=256, N=2048

  for (int t = 0; t < kT; ++t) {
    const float* Xt = Xall + (size_t)t * kB * 32;
    // P1 = [ X_t @ Ui1 | h1 @ Uh1 ]          (256 x 256)
    gemm_wmma_f32<<<g_s1, blk, 0, stream>>>(Xt, 32, Ui1, 128, P1, 256,
                                            kB, 128, 32, nullptr);
    gemm_wmma_f32<<<g_s1, blk, 0, stream>>>(h1, kHID, Uh1, 128, P1 + 128, 256,
                                            kB, 128, kHID, nullptr);
    // gates = P1 @ [Vi1;Vh1] + (bi1+bh1)     (256 x 2048)
    gemm_wmma_f32<<<g_s2, blk, 0, stream>>>(P1, 256, V1cat, kG, gates, kG,
                                            kB, kG, 256, b1cat);
    lstm_update_k<<<kB * kHID / 256, 256, 0, stream>>>(gates, h1, c1);

    // P2 = [ h1 @ Ui2 | h2 @ Uh2 ]
    gemm_wmma_f32<<<g_s1, blk, 0, stream>>>(h1, kHID, Ui2, 128, P2, 256,
                                            kB, 128, kHID, nullptr);
    gemm_wmma_f32<<<g_s1, blk, 0, stream>>>(h2, kHID, Uh2, 128, P2 + 128, 256,
                                            kB, 128, kHID, nullptr);
    gemm_wmma_f32<<<g_s2, blk, 0, stream>>>(P2, 256, V2cat, kG, gates, kG,
                                            kB, kG, 256, b2cat);
    lstm_update_k<<<kB * kHID / 256, 256, 0, stream>>>(gates, h2, c2);
  }

  // ---- final FC: (256,512) @ (512,10)^T + b ------------------------------
  fc_k<<<cdiv(kB * kNC, 256), 256, 0, stream>>>(h2, fcw, fcb, (float*)d_out);
}